// CrossDomainMatching_60172491817603
// MI455X (gfx1250) — compile-verified
//
#include <hip/hip_runtime.h>

// ---------------- problem constants ----------------
#define B_     128
#define L_     4096
#define DIM_   256
#define INNER_ 256
#define SLABS  384          // B_ * 3  (rows of fea)
#define SPLIT  4            // L-dimension split for pooling
#define RPS    (L_ / SPLIT) // rows per split = 1024

typedef float v2f __attribute__((ext_vector_type(2)));
typedef float v4f __attribute__((ext_vector_type(4)));
typedef float v8f __attribute__((ext_vector_type(8)));

// ---------------------------------------------------------------------------
// Kernel 1: bandwidth-bound mean-pool partial sums.
// grid = (SLABS, SPLIT), block = 256.  slab = b*3 + input_index.
// Thread (qid = tid&63, lg = tid>>6) accumulates float4 column qid over rows
// l0+lg, l0+lg+4, ...  Loads are non-temporal (streaming, read-once data that
// is 1.5 GiB >> 192 MB L2) b128s; two accumulators keep 2 independent load
// chains in flight per iteration.
// ---------------------------------------------------------------------------
__global__ void pool_partial_kernel(const float* __restrict__ x1,
                                    const float* __restrict__ x2,
                                    const float* __restrict__ x3,
                                    float* __restrict__ partial) {
  const int slab = blockIdx.x;
  const int sp   = blockIdx.y;
  const int b    = slab / 3;
  const int j    = slab - 3 * b;
  const float* x = (j == 0) ? x1 : (j == 1) ? x2 : x3;
  const v4f* base = (const v4f*)(x + (size_t)b * L_ * DIM_);

  const int tid = threadIdx.x;
  const int qid = tid & 63;   // which float4 within a 256-float row
  const int lg  = tid >> 6;   // 0..3 row-group
  const int l0  = sp * RPS;

  v4f acc0 = {0.f, 0.f, 0.f, 0.f};
  v4f acc1 = {0.f, 0.f, 0.f, 0.f};
#pragma unroll 4
  for (int l = l0 + lg; l < l0 + RPS; l += 8) {
    v4f a = __builtin_nontemporal_load(&base[(size_t)l * 64 + qid]);
    v4f c = __builtin_nontemporal_load(&base[(size_t)(l + 4) * 64 + qid]);
    acc0 += a;
    acc1 += c;
  }
  v4f acc = acc0 + acc1;

  __shared__ v4f red[4][64];
  red[lg][qid] = acc;
  __syncthreads();
  if (lg == 0) {
    v4f s = (red[0][qid] + red[1][qid]) + (red[2][qid] + red[3][qid]);
    ((v4f*)partial)[(size_t)(slab * SPLIT + sp) * 64 + qid] = s;
  }
}

// ---------------------------------------------------------------------------
// Kernel 2: finalize mean + LayerNorm.  grid = SLABS, block = 256 (= DIM).
// Writes fea (residual) and x_ln (normalized input to the QKV projections).
// ---------------------------------------------------------------------------
__global__ void reduce_ln_kernel(const float* __restrict__ partial,
                                 const float* __restrict__ lnw,
                                 const float* __restrict__ lnb,
                                 float* __restrict__ fea,
                                 float* __restrict__ xln) {
  const int slab = blockIdx.x;
  const int d    = threadIdx.x;

  float s = 0.f;
#pragma unroll
  for (int sp = 0; sp < SPLIT; ++sp)
    s += partial[(size_t)(slab * SPLIT + sp) * DIM_ + d];
  const float f = s * (1.0f / L_);
  fea[(size_t)slab * DIM_ + d] = f;

  __shared__ float red[256];
  red[d] = f;
  __syncthreads();
  for (int off = 128; off > 0; off >>= 1) {
    if (d < off) red[d] += red[d + off];
    __syncthreads();
  }
  const float mean = red[0] * (1.0f / DIM_);
  __syncthreads();

  const float cen = f - mean;
  red[d] = cen * cen;
  __syncthreads();
  for (int off = 128; off > 0; off >>= 1) {
    if (d < off) red[d] += red[d + off];
    __syncthreads();
  }
  const float var = red[0] * (1.0f / DIM_);
  const float inv = rsqrtf(var + 1e-5f);
  xln[(size_t)slab * DIM_ + d] = cen * inv * lnw[d] + lnb[d];
}

// ---------------------------------------------------------------------------
// Kernel 3: fp32 WMMA GEMM   Y[r, o] = sum_d X[r, d] * W[o, d] + bias[o]
//           (+ compile-time residual), M = 384, N = 256, K = 256.
// One wave (32 threads) per 16x16 tile, 64 chained V_WMMA_F32_16X16X4_F32.
// A fragment (16x4 f32): lane m = lane&15, half kh = lane>>4;
//   VGPR0/1 = K(2*kh), K(2*kh+1).  B mirrored with n = lane&15.
// C/D: vgpr j holds row (j + 8*kh), column lane&15.
// ---------------------------------------------------------------------------
template <bool HAS_RES>
__global__ void gemm_wmma_kernel(const float* __restrict__ X,
                                 const float* __restrict__ W,
                                 const float* __restrict__ bias,
                                 float* __restrict__ Y,
                                 const float* __restrict__ residual) {
  const int nt   = blockIdx.x;   // 0..15  N tile
  const int mt   = blockIdx.y;   // 0..23  M tile
  const int lane = threadIdx.x;  // 0..31  (one wave32)
  const int mn   = lane & 15;
  const int kh   = lane >> 4;

  const float* __restrict__ xrow = X + (size_t)(mt * 16 + mn) * DIM_;
  const float* __restrict__ wrow = W + (size_t)(nt * 16 + mn) * DIM_;

  v8f c = {0.f, 0.f, 0.f, 0.f, 0.f, 0.f, 0.f, 0.f};
#pragma unroll 8
  for (int kc = 0; kc < DIM_ / 4; ++kc) {
    const int k = kc * 4 + 2 * kh;           // 8-byte aligned
    v2f a  = *(const v2f*)(xrow + k);
    v2f bm = *(const v2f*)(wrow + k);
    c = __builtin_amdgcn_wmma_f32_16x16x4_f32(
        /*neg_a=*/false, a, /*neg_b=*/false, bm,
        /*c_mod=*/(short)0, c, /*reuse_a=*/false, /*reuse_b=*/false);
  }

  const float bv = bias[nt * 16 + mn];
#pragma unroll
  for (int j = 0; j < 8; ++j) {
    const int r = mt * 16 + 8 * kh + j;
    const size_t idx = (size_t)r * INNER_ + nt * 16 + mn;
    float val = c[j] + bv;
    if (HAS_RES) val += residual[idx];
    Y[idx] = val;
  }
}

// ---------------------------------------------------------------------------
// Kernel 4: tiny cross-attention per batch.  grid = B_, block = 256.
// q = heads(x_theta)*SCALE (3 tokens), k/v = heads(x_phi/x_mu) + register
// token (4 tokens).  Outputs attn (b,3,INNER) and head-mean scores (b,3,4).
// ---------------------------------------------------------------------------
__global__ void attention_kernel(const float* __restrict__ xth,
                                 const float* __restrict__ xph,
                                 const float* __restrict__ xmu,
                                 const float* __restrict__ regp,
                                 const float* __restrict__ regm,
                                 float* __restrict__ attn_out,
                                 float* __restrict__ ms_out) {
  const int b   = blockIdx.x;
  const int tid = threadIdx.x;  // 256

  __shared__ float th[3 * 256];
  __shared__ float ph[4 * 256];
  __shared__ float vv[4 * 256];
  __shared__ float gs[48];   // gram,  index h*12 + n*4 + m
  __shared__ float ms[48];   // softmax scores

  for (int i = tid; i < 768; i += 256) {
    th[i] = xth[(size_t)b * 768 + i];
    ph[i] = xph[(size_t)b * 768 + i];
    vv[i] = xmu[(size_t)b * 768 + i];
  }
  ph[768 + tid] = regp[tid];   // register token (row m = 3)
  vv[768 + tid] = regm[tid];
  __syncthreads();

  if (tid < 48) {                       // gram[h][n][m], SCALE = 1/sqrt(64)
    const int h = tid / 12;
    const int rem = tid - 12 * h;
    const int n = rem >> 2;
    const int m = rem & 3;
    float s = 0.f;
#pragma unroll 8
    for (int d = 0; d < 64; ++d)
      s += th[n * 256 + h * 64 + d] * ph[m * 256 + h * 64 + d];
    gs[tid] = s * 0.125f;
  }
  __syncthreads();

  if (tid < 12) {                       // softmax over m (4)
    const int h = tid / 3;
    const int n = tid - 3 * h;
    const int base = h * 12 + n * 4;
    float mx = gs[base];
    for (int m = 1; m < 4; ++m) mx = fmaxf(mx, gs[base + m]);
    float e[4]; float se = 0.f;
    for (int m = 0; m < 4; ++m) { e[m] = __expf(gs[base + m] - mx); se += e[m]; }
    const float inv = 1.f / se;
    for (int m = 0; m < 4; ++m) ms[base + m] = e[m] * inv;
  }
  __syncthreads();

  for (int i = tid; i < 768; i += 256) {   // attn = scores @ V
    const int n = i >> 8;
    const int ccol = i & 255;
    const int h = ccol >> 6;
    const int base = h * 12 + n * 4;
    float s = 0.f;
#pragma unroll
    for (int m = 0; m < 4; ++m) s += ms[base + m] * vv[m * 256 + ccol];
    attn_out[(size_t)b * 768 + i] = s;
  }

  if (tid < 12) {                       // match_score.mean over heads
    const int n = tid >> 2;
    const int m = tid & 3;
    float s = 0.f;
    for (int h = 0; h < 4; ++h) s += ms[h * 12 + n * 4 + m];
    ms_out[(size_t)b * 12 + n * 4 + m] = 0.25f * s;
  }
}

// ---------------------------------------------------------------------------
extern "C" void kernel_launch(void* const* d_in, const int* in_sizes, int n_in,
                              void* d_out, int out_size, void* d_ws, size_t ws_size,
                              hipStream_t stream) {
  const float* x1      = (const float*)d_in[0];
  const float* x2      = (const float*)d_in[1];
  const float* x3      = (const float*)d_in[2];
  const float* ln_w    = (const float*)d_in[3];
  const float* ln_b    = (const float*)d_in[4];
  const float* W_theta = (const float*)d_in[5];
  const float* b_theta = (const float*)d_in[6];
  const float* W_phi   = (const float*)d_in[7];
  const float* b_phi   = (const float*)d_in[8];
  const float* W_mu    = (const float*)d_in[9];
  const float* b_mu    = (const float*)d_in[10];
  const float* W_out   = (const float*)d_in[11];
  const float* b_out   = (const float*)d_in[12];
  const float* reg_phi = (const float*)d_in[13];
  const float* reg_mu  = (const float*)d_in[14];

  float* ws  = (float*)d_ws;
  float* fea = ws;                         // [0,       98304)
  float* xln = ws + 98304;                 // [98304,  196608)
  // partial sums live in [196608, 589824) and are dead after reduce_ln,
  // so the projection/attention buffers alias that region.
  float* partial = ws + 196608;            // 384*4*256 = 393216 floats
  float* xth  = ws + 196608;               // [196608, 294912)
  float* xph  = ws + 294912;               // [294912, 393216)
  float* xmuo = ws + 393216;               // [393216, 491520)
  float* attn = ws + 491520;               // [491520, 589824)

  float* out = (float*)d_out;              // 98304 out + 1536 match-score

  dim3 gpool(SLABS, SPLIT);
  pool_partial_kernel<<<gpool, 256, 0, stream>>>(x1, x2, x3, partial);

  reduce_ln_kernel<<<SLABS, 256, 0, stream>>>(partial, ln_w, ln_b, fea, xln);

  dim3 gg(INNER_ / 16, SLABS / 16);        // (16, 24) tiles, 1 wave each
  gemm_wmma_kernel<false><<<gg, 32, 0, stream>>>(xln, W_theta, b_theta, xth,  nullptr);
  gemm_wmma_kernel<false><<<gg, 32, 0, stream>>>(xln, W_phi,   b_phi,   xph,  nullptr);
  gemm_wmma_kernel<false><<<gg, 32, 0, stream>>>(xln, W_mu,    b_mu,    xmuo, nullptr);

  attention_kernel<<<B_, 256, 0, stream>>>(xth, xph, xmuo, reg_phi, reg_mu,
                                           attn, out + 98304);

  // final projection + bias + residual(fea), written straight into d_out
  gemm_wmma_kernel<true><<<gg, 32, 0, stream>>>(attn, W_out, b_out, out, fea);
}